// BiConvLSTM_47510928228795
// MI455X (gfx1250) — compile-verified
//
#include <hip/hip_runtime.h>
#include <cmath>

typedef float v2f __attribute__((ext_vector_type(2)));
typedef float v8f __attribute__((ext_vector_type(8)));

#define HW    2304
#define G     9216          // 4*HW
#define T_    16
#define NCOL  16            // combined fwd(8) + bwd(8) columns
#define KCH   576           // K-chunk staged in LDS
#define LDSROW 580          // 576+4 -> bank-conflict-free (576 % 64 == 0)

// workspace layout (floats)
#define XP_OFF  0                   // Xp[T][G][8]  (x-projection + biases)
#define XP_SZ   (T_ * G * 8)        // 1,179,648
#define GT_OFF  (XP_OFF + XP_SZ)    // gates[G][16]
#define GT_SZ   (G * NCOL)          // 147,456
#define H_OFF   (GT_OFF + GT_SZ)    // Hcat[16][HW]
#define H_SZ    (NCOL * HW)
#define C_OFF   (H_OFF + H_SZ)      // Ccat[16][HW]
#define C_SZ    (NCOL * HW)

__device__ __forceinline__ float sigmoidf_(float v) { return 1.f / (1.f + __expf(-v)); }

__global__ void k_zero(float* p, int n) {
    int i = blockIdx.x * blockDim.x + threadIdx.x;
    if (i < n) p[i] = 0.f;
}

// ---------------------------------------------------------------------------
// Phase 1: Xp[t][g][b] = sum_k W_ih[g,k] * x[b,t,k] + b_ih[g] + b_hh[g]
// One [9216 x 2304] x [2304 x 128] WMMA-f32 GEMM. 4608 tiles, 1 tile/wave.
// ---------------------------------------------------------------------------
__global__ void k_xproj(const float* __restrict__ x,
                        const float* __restrict__ Wih,
                        const float* __restrict__ bih,
                        const float* __restrict__ bhh,
                        float* __restrict__ Xp) {
    const int lane = threadIdx.x & 31;
    const int wid  = threadIdx.x >> 5;
    const int wg   = blockIdx.x * (blockDim.x >> 5) + wid;  // 0..4607
    // N-tile fast-varying so consecutive blocks reuse W_ih rows via L2
    const int mtile = wg >> 3;               // 0..575
    const int ntile = wg & 7;                // 0..7
    const int n  = lane & 15;                // col in tile / A row (M=lane&15)
    const int hh = lane >> 4;                // K half: K = 2*hh + v
    const int g0 = mtile * 16;
    const int ncol = ntile * 16 + n;         // 0..127
    const int t = ncol >> 3;
    const int b = ncol & 7;

    const float* Arow = Wih + (size_t)(g0 + n) * HW + 2 * hh;
    const float* Brow = x + (size_t)b * (T_ * HW) + (size_t)t * HW + 2 * hh;

    v8f c0 = {}, c1 = {};
    for (int k = 0; k < HW; k += 8) {
        v2f a0 = *(const v2f*)(Arow + k);
        v2f b0 = *(const v2f*)(Brow + k);
        v2f a1 = *(const v2f*)(Arow + k + 4);
        v2f b1 = *(const v2f*)(Brow + k + 4);
        c0 = __builtin_amdgcn_wmma_f32_16x16x4_f32(false, a0, false, b0, (short)0, c0, false, false);
        c1 = __builtin_amdgcn_wmma_f32_16x16x4_f32(false, a1, false, b1, (short)0, c1, false, false);
    }
#pragma unroll
    for (int r = 0; r < 8; ++r) {
        int g = g0 + r + 8 * hh;            // D: M = r + 8*(lane>=16), N = lane&15
        float v = c0[r] + c1[r] + bih[g] + bhh[g];
        Xp[(size_t)t * (G * 8) + (size_t)g * 8 + b] = v;
    }
}

// ---------------------------------------------------------------------------
// Phase 2a (x16): gates[g][j] = Xp(+biases) + sum_k W_hh[g,k] * Hcat[j,k]
// j: 0-7 forward batch (time t), 8-15 backward batch (time T-1-t).
// W_hh streams from L2; Hcat (B operand, same for every tile) staged in LDS.
// ---------------------------------------------------------------------------
__global__ void k_step(const float* __restrict__ Whh,
                       const float* __restrict__ Xp,
                       const float* __restrict__ Hcat,
                       float* __restrict__ gates,
                       int t) {
    __shared__ float lds[NCOL * LDSROW];     // 37,120 B, bank-conflict-free rows

    const int lane = threadIdx.x & 31;
    const int wid  = threadIdx.x >> 5;       // 0..7
    const int n  = lane & 15;
    const int hh = lane >> 4;
    const int tile = blockIdx.x * 8 + wid;   // 0..575
    const int g0 = tile * 16;

    const float* Arow = Whh + (size_t)(g0 + n) * HW + 2 * hh;

    // init accumulator from precomputed x-projection (+ both biases)
    const int tb   = (n < 8) ? t : (T_ - 1 - t);
    const int bcol = n & 7;
    const float* Xpt = Xp + (size_t)tb * (G * 8);
    v8f c0, c1 = {};
#pragma unroll
    for (int r = 0; r < 8; ++r)
        c0[r] = Xpt[(size_t)(g0 + r + 8 * hh) * 8 + bcol];

    for (int kc = 0; kc < HW; kc += KCH) {
        __syncthreads();                     // previous chunk fully consumed
        for (int i = threadIdx.x; i < NCOL * KCH; i += blockDim.x) {
            int row = i / KCH, col = i - row * KCH;
            lds[row * LDSROW + col] = Hcat[row * HW + kc + col];
        }
        __syncthreads();
        const float* Bp = lds + n * LDSROW + 2 * hh;
#pragma unroll 4
        for (int k = 0; k < KCH; k += 8) {
            v2f b0 = *(const v2f*)(Bp + k);
            v2f b1 = *(const v2f*)(Bp + k + 4);
            v2f a0 = *(const v2f*)(Arow + kc + k);
            v2f a1 = *(const v2f*)(Arow + kc + k + 4);
            c0 = __builtin_amdgcn_wmma_f32_16x16x4_f32(false, a0, false, b0, (short)0, c0, false, false);
            c1 = __builtin_amdgcn_wmma_f32_16x16x4_f32(false, a1, false, b1, (short)0, c1, false, false);
        }
    }
#pragma unroll
    for (int r = 0; r < 8; ++r)
        gates[(size_t)(g0 + r + 8 * hh) * 16 + n] = c0[r] + c1[r];
}

// ---------------------------------------------------------------------------
// Phase 2b (x16): LSTM cell elementwise. Gate order i,f,g,o over 4*HW rows.
// ---------------------------------------------------------------------------
__global__ void k_cell(const float* __restrict__ gates,
                       float* __restrict__ Hcat,
                       float* __restrict__ Ccat) {
    int idx = blockIdx.x * blockDim.x + threadIdx.x;
    if (idx >= NCOL * HW) return;
    int j = idx & 15;                        // column (coalesced gate reads)
    int u = idx >> 4;                        // hidden index
    float gi = gates[(size_t)u * 16 + j];
    float gf = gates[(size_t)(u + HW) * 16 + j];
    float gg = gates[(size_t)(u + 2 * HW) * 16 + j];
    float go = gates[(size_t)(u + 3 * HW) * 16 + j];
    float cp = Ccat[j * HW + u];
    float cn = sigmoidf_(gf) * cp + sigmoidf_(gi) * tanhf(gg);
    Ccat[j * HW + u] = cn;
    Hcat[j * HW + u] = sigmoidf_(go) * tanhf(cn);
}

// ---------------------------------------------------------------------------
// Phase 3: faithful flat-reshape + tanh + 1x1 conv (2ch -> 1) + LeakyReLU.
// out[b'][u] mixes combined rows 2b' and 2b'+1 of [h_fwd; h_bwd].
// ---------------------------------------------------------------------------
__global__ void k_out(const float* __restrict__ Hcat,
                      const float* __restrict__ cw,
                      const float* __restrict__ cb,
                      float* __restrict__ out) {
    int idx = blockIdx.x * blockDim.x + threadIdx.x;
    if (idx >= 8 * HW) return;
    int bp = idx / HW, u = idx - bp * HW;
    float h0 = Hcat[(2 * bp) * HW + u];
    float h1 = Hcat[(2 * bp + 1) * HW + u];
    float y = cw[0] * tanhf(h0) + cw[1] * tanhf(h1) + cb[0];
    out[idx] = (y >= 0.f) ? y : 0.01f * y;
}

extern "C" void kernel_launch(void* const* d_in, const int* in_sizes, int n_in,
                              void* d_out, int out_size, void* d_ws, size_t ws_size,
                              hipStream_t stream) {
    const float* x   = (const float*)d_in[0];
    const float* Wih = (const float*)d_in[1];
    const float* Whh = (const float*)d_in[2];
    const float* bih = (const float*)d_in[3];
    const float* bhh = (const float*)d_in[4];
    const float* cw  = (const float*)d_in[5];
    const float* cb  = (const float*)d_in[6];
    float* out = (float*)d_out;
    float* ws  = (float*)d_ws;

    float* Xp    = ws + XP_OFF;
    float* gates = ws + GT_OFF;
    float* Hc    = ws + H_OFF;
    float* Cc    = ws + C_OFF;

    // h0 = c0 = 0 (Hcat and Ccat are contiguous in workspace)
    k_zero<<<(H_SZ + C_SZ + 255) / 256, 256, 0, stream>>>(Hc, H_SZ + C_SZ);

    // one-shot input projection GEMM: 4608 tiles, 4 waves/block
    k_xproj<<<1152, 128, 0, stream>>>(x, Wih, bih, bhh, Xp);

    // 16 sequential steps, fwd+bwd fused into one 16-column GEMM
    for (int t = 0; t < T_; ++t) {
        k_step<<<72, 256, 0, stream>>>(Whh, Xp, Hc, gates, t);
        k_cell<<<(NCOL * HW + 255) / 256, 256, 0, stream>>>(gates, Hc, Cc);
    }

    k_out<<<(8 * HW + 255) / 256, 256, 0, stream>>>(Hc, cw, cb, out);
}